// CircleLoss_31885837205664
// MI455X (gfx1250) — compile-verified
//
#include <hip/hip_runtime.h>
#include <math.h>

typedef __attribute__((ext_vector_type(2))) float v2f;
typedef __attribute__((ext_vector_type(8))) float v8f;

#define NB 4096          // batch
#define DK 512           // embedding dim
#define NCHUNK 8         // column chunks (parallel over grid.y)
#define COLS_PER_CHUNK (NB / NCHUNK)          // 512
#define TILES_PER_CHUNK (COLS_PER_CHUNK / 16) // 32
#define NFIELD 10        // per-(row,chunk) partial stats

// ---- streaming logsumexp helpers (m = running max, s = sum of exp(x-m)) ----
__device__ __forceinline__ void lse_update(float& m, float& s, float x) {
  if (x > m) { s = s * __expf(m - x) + 1.0f; m = x; }  // expf(-inf)=0 handles first elem
  else       { s += __expf(x - m); }
}

__device__ __forceinline__ void lse_merge(float& m, float& s, float m2, float s2) {
  float mm = fmaxf(m, m2);
  if (mm > -INFINITY) s = s * __expf(m - mm) + s2 * __expf(m2 - mm);
  m = mm;
}

// ======================= Phase 1: fused GEMM + row stats ====================
__global__ __launch_bounds__(32)
void circle_phase1(const float* __restrict__ E, const int* __restrict__ labels,
                   float* __restrict__ ws) {
  __shared__ float Alds[16 * DK];   // 32 KB: this block's 16 rows
  __shared__ int   labR[16];

  const int lane  = threadIdx.x;    // wave32
  const int rt    = blockIdx.x;     // row tile 0..255
  const int chunk = blockIdx.y;     // column chunk 0..7
  const int half  = lane >> 4;      // 0: k0..k0+1, 1: k0+2..k0+3
  const int l15   = lane & 15;

  // Stage A rows (contiguous in global) into LDS, coalesced b128.
  {
    const float4* src = (const float4*)(E + (size_t)rt * 16 * DK);
    float4*       dst = (float4*)Alds;
    #pragma unroll 8
    for (int i = 0; i < (16 * DK / 4) / 32; ++i)
      dst[lane + i * 32] = src[lane + i * 32];
    if (lane < 16) labR[lane] = labels[rt * 16 + lane];
  }
  __syncthreads();

  int labrow[8];
  #pragma unroll
  for (int v = 0; v < 8; ++v) labrow[v] = labR[v + 8 * half];

  // Per-lane stats for 8 rows (C-matrix layout: lane -> column N=l15, rows v+8*half)
  float mp1[8], sp1[8], mp2[8], sp2[8], mn1[8], sn1[8], mn2[8], sn2[8], pmin[8], nmax[8];
  #pragma unroll
  for (int v = 0; v < 8; ++v) {
    mp1[v] = -INFINITY; sp1[v] = 0.f;
    mp2[v] = -INFINITY; sp2[v] = 0.f;
    mn1[v] = -INFINITY; sn1[v] = 0.f;
    mn2[v] = -INFINITY; sn2[v] = 0.f;
    pmin[v] = INFINITY; nmax[v] = -INFINITY;
  }

  const int koff = half * 2;
  const float* Arow = Alds + l15 * DK + koff;   // A frag source (LDS)

  #pragma unroll 1
  for (int ct = 0; ct < TILES_PER_CHUNK; ++ct) {
    const int col_base = chunk * COLS_PER_CHUNK + ct * 16;
    const float* Brow = E + (size_t)(col_base + l15) * DK + koff;  // B frag (global, L2-hot)

    v8f acc0 = {}; v8f acc1 = {};   // two accumulators to break the WMMA D->C chain
    #pragma unroll 4
    for (int k0 = 0; k0 < DK; k0 += 8) {
      v2f a0 = *(const v2f*)(Arow + k0);
      v2f b0 = *(const v2f*)(Brow + k0);
      acc0 = __builtin_amdgcn_wmma_f32_16x16x4_f32(false, a0, false, b0,
                                                   (short)0, acc0, false, false);
      v2f a1 = *(const v2f*)(Arow + k0 + 4);
      v2f b1 = *(const v2f*)(Brow + k0 + 4);
      acc1 = __builtin_amdgcn_wmma_f32_16x16x4_f32(false, a1, false, b1,
                                                   (short)0, acc1, false, false);
    }
    v8f acc = acc0 + acc1;

    const int labC = labels[col_base + l15];
    #pragma unroll
    for (int v = 0; v < 8; ++v) {
      float s = acc[v];
      if (labC == labrow[v]) {                    // positive pair (incl. diagonal)
        pmin[v] = fminf(pmin[v], s);
        float ip = -32.0f * fmaxf(1.25f - s, 0.0f) * (s - 0.75f);
        if (s < 0.5f)  lse_update(mp1[v], sp1[v], ip);   // mask_p1
        if (s < 0.75f) lse_update(mp2[v], sp2[v], ip);   // mask_p2
      } else {                                    // negative pair
        nmax[v] = fmaxf(nmax[v], s);
        float in = 32.0f * fmaxf(s + 0.25f, 0.0f) * (s - 0.25f);
        if (s > 0.5f)  lse_update(mn1[v], sn1[v], in);   // mask_n1
        if (s > 0.25f) lse_update(mn2[v], sn2[v], in);   // mask_n2
      }
    }
  }

  // Merge stats across the 16 lanes that share each row (xor never crosses halves).
  #pragma unroll
  for (int off = 1; off < 16; off <<= 1) {
    #pragma unroll
    for (int v = 0; v < 8; ++v) {
      { float om = __shfl_xor(mp1[v], off), os = __shfl_xor(sp1[v], off); lse_merge(mp1[v], sp1[v], om, os); }
      { float om = __shfl_xor(mp2[v], off), os = __shfl_xor(sp2[v], off); lse_merge(mp2[v], sp2[v], om, os); }
      { float om = __shfl_xor(mn1[v], off), os = __shfl_xor(sn1[v], off); lse_merge(mn1[v], sn1[v], om, os); }
      { float om = __shfl_xor(mn2[v], off), os = __shfl_xor(sn2[v], off); lse_merge(mn2[v], sn2[v], om, os); }
      pmin[v] = fminf(pmin[v], __shfl_xor(pmin[v], off));
      nmax[v] = fmaxf(nmax[v], __shfl_xor(nmax[v], off));
    }
  }

  // Lane (l15==v) in each half writes its row's partial stats.
  #pragma unroll
  for (int v = 0; v < 8; ++v) {
    if (l15 == v) {
      int row = rt * 16 + v + 8 * half;
      float* p = ws + ((size_t)row * NCHUNK + chunk) * NFIELD;
      p[0] = mp1[v]; p[1] = sp1[v];
      p[2] = mp2[v]; p[3] = sp2[v];
      p[4] = mn1[v]; p[5] = sn1[v];
      p[6] = mn2[v]; p[7] = sn2[v];
      p[8] = pmin[v]; p[9] = nmax[v];
    }
  }
}

// =================== Phase 2: per-row finalize + global reduce ==============
__global__ __launch_bounds__(256)
void circle_phase2(const float* __restrict__ ws, float* __restrict__ out) {
  const int row = blockIdx.x * blockDim.x + threadIdx.x;

  float mp1 = -INFINITY, sp1 = 0.f, mp2 = -INFINITY, sp2 = 0.f;
  float mn1 = -INFINITY, sn1 = 0.f, mn2 = -INFINITY, sn2 = 0.f;
  float pmin = INFINITY, nmax = -INFINITY;
  #pragma unroll
  for (int c = 0; c < NCHUNK; ++c) {
    const float* p = ws + ((size_t)row * NCHUNK + c) * NFIELD;
    lse_merge(mp1, sp1, p[0], p[1]);
    lse_merge(mp2, sp2, p[2], p[3]);
    lse_merge(mn1, sn1, p[4], p[5]);
    lse_merge(mn2, sn2, p[6], p[7]);
    pmin = fminf(pmin, p[8]);
    nmax = fmaxf(nmax, p[9]);
  }

  float fbp = -32.0f * fmaxf(1.25f - pmin, 0.0f) * (pmin - 0.75f);
  float fbn =  32.0f * fmaxf(nmax + 0.25f, 0.0f) * (nmax - 0.25f);
  float se_pos = (sp1 > 0.f) ? (mp1 + __logf(sp1))
               : ((sp2 > 0.f) ? (mp2 + __logf(sp2)) : fbp);
  float se_neg = (sn1 > 0.f) ? (mn1 + __logf(sn1))
               : ((sn2 > 0.f) ? (mn2 + __logf(sn2)) : fbn);
  float z = se_pos + se_neg;
  float spl = fmaxf(z, 0.0f) + log1pf(__expf(-fabsf(z)));  // stable softplus

  float low  = (pmin > nmax) ? 1.f : 0.f;
  float med  = (low  > 0.f && pmin > 0.75f) ? 1.f : 0.f;
  float high = (med  > 0.f && nmax < 0.25f) ? 1.f : 0.f;

  const float inv = 1.0f / (float)NB;
  float c0 = spl * inv, c1 = low * inv, c2 = med * inv, c3 = high * inv;
  #pragma unroll
  for (int off = 16; off > 0; off >>= 1) {
    c0 += __shfl_down(c0, off);
    c1 += __shfl_down(c1, off);
    c2 += __shfl_down(c2, off);
    c3 += __shfl_down(c3, off);
  }
  if ((threadIdx.x & 31) == 0) {
    atomicAdd(&out[0], c0);
    atomicAdd(&out[1], c1);
    atomicAdd(&out[2], c2);
    atomicAdd(&out[3], c3);
  }
}

__global__ void circle_init(float* __restrict__ out) {
  if (threadIdx.x < 4) out[threadIdx.x] = 0.0f;
}

extern "C" void kernel_launch(void* const* d_in, const int* in_sizes, int n_in,
                              void* d_out, int out_size, void* d_ws, size_t ws_size,
                              hipStream_t stream) {
  (void)in_sizes; (void)n_in; (void)out_size; (void)ws_size;
  const float* E      = (const float*)d_in[0];
  const int*   labels = (const int*)d_in[1];
  float*       out    = (float*)d_out;
  float*       ws     = (float*)d_ws;   // needs NB*NCHUNK*NFIELD floats = 1.25 MB

  circle_init<<<1, 32, 0, stream>>>(out);
  circle_phase1<<<dim3(256, NCHUNK), 32, 0, stream>>>(E, labels, ws);
  circle_phase2<<<NB / 256, 256, 0, stream>>>(ws, out);
}